// AdaptiveGraphConvolution_12592844112362
// MI455X (gfx1250) — compile-verified
//
#include <hip/hip_runtime.h>
#include <hip/hip_bf16.h>

typedef float v2f __attribute__((ext_vector_type(2)));
typedef float v8f __attribute__((ext_vector_type(8)));

#define IN_DIM 256
#define OUT_DIM 128
#define LDX 260   // padded LDS row stride (floats): 260 % 64 == 4 -> conflict-free b64 A-frag loads

// ---------------------------------------------------------------------------
// Kernel 1: fused dual GEMM  pre = x@W, fpre = x@fsW  via V_WMMA_F32_16X16X4_F32
// Block = 256 threads = 8 waves. Block computes a 16-row stripe of both outputs;
// wave w owns the 16-wide column tile [16w, 16w+16).
// ---------------------------------------------------------------------------
__global__ __launch_bounds__(256) void agc_gemm_dual(
    const float* __restrict__ x, const float* __restrict__ W,
    const float* __restrict__ fsW, float* __restrict__ pre,
    float* __restrict__ fpre, int nNodes) {
  __shared__ float xs[16 * LDX];

  const int t  = threadIdx.x;
  const int rb = blockIdx.x * 16;

  // Stage 16x256 x-tile into LDS: 4096 floats = 1024 float4, 4 per thread.
  #pragma unroll
  for (int j = 0; j < 4; ++j) {
    int i  = t + 256 * j;       // float4 index
    int r  = i >> 6;            // row in tile (64 float4 per row)
    int c4 = i & 63;
    int gr = rb + r;
    if (gr >= nNodes) gr = nNodes - 1;   // clamp (keeps EXEC uniform)
    float4 v = *(const float4*)(x + (size_t)gr * IN_DIM + (size_t)c4 * 4);
    *(float4*)(xs + r * LDX + c4 * 4) = v;
  }
  __syncthreads();

  const int wave = t >> 5;
  const int lane = t & 31;
  const int hi   = lane >> 4;   // 0: K pair {k,k+1}, 1: K pair {k+2,k+3}
  const int ln   = lane & 15;   // M for A-frag, N for B/C/D frags
  const int cb   = wave * 16;   // column tile base

  v8f accA = {0.f,0.f,0.f,0.f,0.f,0.f,0.f,0.f};
  v8f accB = {0.f,0.f,0.f,0.f,0.f,0.f,0.f,0.f};

  const float* wA = W   + cb + ln;   // column pointers
  const float* wB = fsW + cb + ln;
  const float* as = xs + ln * LDX + 2 * hi;

  #pragma unroll 8
  for (int k = 0; k < IN_DIM; k += 4) {
    const int ka = k + 2 * hi;
    // A-frag: ISA layout lanes0-15 -> A[M=ln][ka], A[ln][ka+1]; lanes16-31 -> K+2
    v2f a = *(const v2f*)(as + k);
    // B-frags: lane holds B[ka][cb+ln], B[ka+1][cb+ln]
    v2f bA, bB;
    bA[0] = wA[(size_t)ka * OUT_DIM];
    bA[1] = wA[(size_t)(ka + 1) * OUT_DIM];
    bB[0] = wB[(size_t)ka * OUT_DIM];
    bB[1] = wB[(size_t)(ka + 1) * OUT_DIM];
    accA = __builtin_amdgcn_wmma_f32_16x16x4_f32(false, a, false, bA,
                                                 (short)0, accA, false, false);
    accB = __builtin_amdgcn_wmma_f32_16x16x4_f32(false, a, false, bB,
                                                 (short)0, accB, false, false);
  }

  // D layout: VGPR v holds row M = v + 8*hi, col N = ln
  #pragma unroll
  for (int v = 0; v < 8; ++v) {
    int gr = rb + v + 8 * hi;
    if (gr < nNodes) {
      size_t o = (size_t)gr * OUT_DIM + cb + ln;
      pre[o]  = accA[v];
      fpre[o] = accB[v];
    }
  }
}

// ---------------------------------------------------------------------------
// Kernel 2: per-node scores  si[n] = fpre[n,:]·wi,  sj[n] = fpre[n,:]·wj
// One wave32 per node; lane handles 4 contiguous elements; shfl_xor reduce.
// ---------------------------------------------------------------------------
__global__ __launch_bounds__(256) void agc_node_scores(
    const float* __restrict__ fpre, const float* __restrict__ fw,
    float* __restrict__ si, float* __restrict__ sj, int nNodes) {
  const int wave = threadIdx.x >> 5;
  const int lane = threadIdx.x & 31;
  const int node = blockIdx.x * 8 + wave;
  if (node >= nNodes) return;

  float4 v = *(const float4*)(fpre + (size_t)node * OUT_DIM + lane * 4);
  float4 a = *(const float4*)(fw + lane * 4);            // wi = fw[0:128]
  float4 b = *(const float4*)(fw + OUT_DIM + lane * 4);  // wj = fw[128:256]
  float di = v.x * a.x + v.y * a.y + v.z * a.z + v.w * a.w;
  float dj = v.x * b.x + v.y * b.y + v.z * b.z + v.w * b.w;
  #pragma unroll
  for (int m = 16; m > 0; m >>= 1) {
    di += __shfl_xor(di, m, 32);
    dj += __shfl_xor(dj, m, 32);
  }
  if (lane == 0) { si[node] = di; sj[node] = dj; }
}

// ---------------------------------------------------------------------------
// Kernel 3: zero the output accumulator (float4 stores)
// ---------------------------------------------------------------------------
__global__ __launch_bounds__(256) void agc_zero(float4* __restrict__ out, int n4) {
  int i = blockIdx.x * blockDim.x + threadIdx.x;
  if (i < n4) out[i] = make_float4(0.f, 0.f, 0.f, 0.f);
}

// ---------------------------------------------------------------------------
// Kernel 4: edge scatter.  One wave32 per edge; lane covers 4 output channels.
// val = (deg_r*deg_c)^(-sigmoid(si[r]+sj[c]+b)); out[r] += val * pre[c]
// ---------------------------------------------------------------------------
__global__ __launch_bounds__(256) void agc_edge_scatter(
    const int* __restrict__ row, const int* __restrict__ col,
    const float* __restrict__ si, const float* __restrict__ sj,
    const float* __restrict__ fb, const float* __restrict__ dia,
    const float* __restrict__ pre, float* __restrict__ out, int nEdges) {
  const int wave = threadIdx.x >> 5;
  const int lane = threadIdx.x & 31;
  const int e = blockIdx.x * 8 + wave;
  if (e >= nEdges) return;

  const int r = row[e];
  const int c = col[e];
  const float z     = si[r] + sj[c] + fb[0];
  const float score = 1.0f / (1.0f + __expf(-z));       // sigmoid
  const float deg   = dia[r] * dia[c];                  // in [1, 4096] -> log ok
  const float val   = __expf(-score * __logf(deg));     // deg^(-score)

  const float4 p = *(const float4*)(pre + (size_t)c * OUT_DIM + lane * 4);
  float* o = out + (size_t)r * OUT_DIM + lane * 4;
  atomicAdd(o + 0, val * p.x);
  atomicAdd(o + 1, val * p.y);
  atomicAdd(o + 2, val * p.z);
  atomicAdd(o + 3, val * p.w);
}

// ---------------------------------------------------------------------------
// Kernel 5: ReLU in place
// ---------------------------------------------------------------------------
__global__ __launch_bounds__(256) void agc_relu(float* __restrict__ out, int n) {
  int i = blockIdx.x * blockDim.x + threadIdx.x;
  if (i < n) out[i] = fmaxf(out[i], 0.f);
}

// ---------------------------------------------------------------------------
extern "C" void kernel_launch(void* const* d_in, const int* in_sizes, int n_in,
                              void* d_out, int out_size, void* d_ws, size_t ws_size,
                              hipStream_t stream) {
  const float* x   = (const float*)d_in[0];
  const float* W   = (const float*)d_in[1];
  const float* fsW = (const float*)d_in[2];
  const float* fw  = (const float*)d_in[3];
  const float* fb  = (const float*)d_in[4];
  const float* dia = (const float*)d_in[5];
  const int*   row = (const int*)d_in[6];
  const int*   col = (const int*)d_in[7];
  float* out = (float*)d_out;

  const int nNodes = in_sizes[5];   // dia_adj length
  const int nEdges = in_sizes[6];   // row length

  // Workspace layout: pre | fpre | si | sj
  float* pre  = (float*)d_ws;
  float* fpre = pre  + (size_t)nNodes * OUT_DIM;
  float* si   = fpre + (size_t)nNodes * OUT_DIM;
  float* sj   = si   + nNodes;

  // 1) zero output accumulator
  const int n4 = out_size / 4;
  agc_zero<<<(n4 + 255) / 256, 256, 0, stream>>>((float4*)out, n4);

  // 2) fused dual GEMM (WMMA f32 16x16x4)
  const int gemmBlocks = (nNodes + 15) / 16;
  agc_gemm_dual<<<gemmBlocks, 256, 0, stream>>>(x, W, fsW, pre, fpre, nNodes);

  // 3) per-node attention scalars
  agc_node_scores<<<(nNodes + 7) / 8, 256, 0, stream>>>(fpre, fw, si, sj, nNodes);

  // 4) edge gather / atomic scatter (bandwidth-dominant phase, L2-resident)
  agc_edge_scatter<<<(nEdges + 7) / 8, 256, 0, stream>>>(row, col, si, sj, fb,
                                                         dia, pre, out, nEdges);

  // 5) ReLU
  agc_relu<<<(out_size + 255) / 256, 256, 0, stream>>>(out, out_size);
}